// SiTMAEEmbeddings_89799176225214
// MI455X (gfx1250) — compile-verified
//
#include <hip/hip_runtime.h>
#include <math.h>

#define Bn    16
#define Ln    4096
#define PDn   512
#define Hn    256
#define FREQn 256

typedef __attribute__((ext_vector_type(2))) float v2f;
typedef __attribute__((ext_vector_type(8))) float v8f;

// ---------------------------------------------------------------------------
// Kernel W-transpose (one-time, 512 KB), K-pair interleaved:
//   WtG2[(k/2)*Hn*2 + n*2 + (k&1)] = W[n][k]
// A B-fragment (B(k,n), B(k+1,n)) is then 8 contiguous bytes, so the WMMA
// B operand is a single aligned ds_load_b64 (no VGPR repacking), and each
// pair-row (512 B) stays contiguous for async global->LDS b128 copies.
// ---------------------------------------------------------------------------
__global__ __launch_bounds__(256)
void k_wtr(const float* __restrict__ W, float* __restrict__ WtG2)
{
    const int k = blockIdx.x;       // 0..511
    const int n = threadIdx.x;      // 0..255
    WtG2[((size_t)(k >> 1) * Hn + n) * 2 + (k & 1)] = W[(size_t)n * PDn + k];
}

// ---------------------------------------------------------------------------
// Kernel A: emb[row, h] = sum_p patches[row,p] * W[h,p] + bias[h] + pos[l,h]
// f32 WMMA 16x16x4. Each wave: 16 rows x 64 cols (4 accumulators), A fragment
// reused across the 4 n-tiles. W streamed K-chunk-wise (128 k = 64 pairs)
// through a double-buffered LDS pipeline fed by async global->LDS loads.
// ---------------------------------------------------------------------------
__global__ __launch_bounds__(128)
void k_proj(const float* __restrict__ patches, const float* __restrict__ WtG2,
            const float* __restrict__ bias, const float* __restrict__ pos,
            float* __restrict__ emb)
{
    __shared__ float Wt[2][64 * 128];       // two 32 KB chunk buffers [pair][n][2]

    const int n0     = blockIdx.x * 64;     // 64-wide output column strip
    const int row0wg = blockIdx.y * 64;     // 4 row-tiles per workgroup
    const int tid    = threadIdx.x;

    const int lane  = tid & 31;
    const int wave  = tid >> 5;
    const int row0  = row0wg + wave * 16;   // this wave's 16 output rows
    const int nlo   = lane & 15;
    const int plane = lane >> 4;            // lanes 0-15 -> K{0,1}, 16-31 -> K{2,3}

    // Async-copy pattern per 64-pair chunk (16 b128 ops per thread):
    // thread t covers pair-rows prow + 4*i (i=0..15), 16-B group quad in row.
    const int quad = tid & 31;              // 0..31 (32 x 16 B = 512 B row)
    const int prow = tid >> 5;              // 0..3

    // ---- prologue: kick off chunk 0 into buffer 0 ----
    {
        unsigned lds0 = (unsigned)(size_t)&Wt[0][prow * 128 + quad * 4];
        const float* g0 = WtG2 + ((size_t)prow * Hn + n0) * 2 + quad * 4;
        #pragma unroll
        for (int i = 0; i < 16; ++i) {
            asm volatile("global_load_async_to_lds_b128 %0, %1, off"
                         :: "v"(lds0 + (unsigned)(i * 2048)),   // 4 rows * 512 B
                            "v"(g0 + (size_t)i * 4 * Hn * 2)
                         : "memory");
        }
    }

    const float* aptr = patches + (size_t)(row0 + nlo) * PDn + plane * 2;

    v8f acc[4];
    #pragma unroll
    for (int nt = 0; nt < 4; ++nt)
        acc[nt] = (v8f){0.f, 0.f, 0.f, 0.f, 0.f, 0.f, 0.f, 0.f};

    const int NCHUNK = PDn / 128;           // 4
    for (int ch = 0; ch < NCHUNK; ++ch) {
        if (ch + 1 < NCHUNK) {
            // issue next chunk into the other buffer
            unsigned lds0 = (unsigned)(size_t)&Wt[(ch + 1) & 1][prow * 128 + quad * 4];
            const float* g0 = WtG2 + ((size_t)((ch + 1) * 64 + prow) * Hn + n0) * 2 + quad * 4;
            #pragma unroll
            for (int i = 0; i < 16; ++i) {
                asm volatile("global_load_async_to_lds_b128 %0, %1, off"
                             :: "v"(lds0 + (unsigned)(i * 2048)),
                                "v"(g0 + (size_t)i * 4 * Hn * 2)
                             : "memory");
            }
            // drain everything except the 16 just-issued loads -> chunk ch ready
            asm volatile("s_wait_asynccnt 0x10" ::: "memory");
            __builtin_prefetch(aptr + ch * 128 + 128, 0, 1);    // A stream ahead
        } else {
            asm volatile("s_wait_asynccnt 0x0" ::: "memory");
        }
        __syncthreads();                    // publish chunk ch to all waves

        const float* bufp = &Wt[ch & 1][0];
        const int kb0 = ch * 128;
        for (int ks = 0; ks < 128; ks += 4) {
            v2f a = *(const v2f*)(aptr + kb0 + ks);             // 8-byte A load
            // pair index in chunk = ks/2 + plane; float offset = pair*128 + n*2
            const float* bp = bufp + (size_t)((ks >> 1) + plane) * 128 + nlo * 2;
            #pragma unroll
            for (int nt = 0; nt < 4; ++nt) {
                v2f bf = *(const v2f*)(bp + nt * 32);           // (B(k,n),B(k+1,n))
                acc[nt] = __builtin_amdgcn_wmma_f32_16x16x4_f32(
                        /*neg_a=*/false, a, /*neg_b=*/false, bf,
                        /*c_mod=*/(short)0, acc[nt],
                        /*reuse_a=*/false, /*reuse_b=*/false);
            }
        }
        __syncthreads();                    // buffer may be overwritten next iter
    }

    // Epilogue: D(m,n) -> reg r, lanes0-15: M=r, lanes16-31: M=r+8
    const int mhi = plane * 8;
    #pragma unroll
    for (int nt = 0; nt < 4; ++nt) {
        const int n  = n0 + nt * 16 + nlo;
        const float bn = bias[n];
        #pragma unroll
        for (int r = 0; r < 8; ++r) {
            const int row = row0 + mhi + r;
            const int l   = row & (Ln - 1);     // row = b*L + l
            emb[(size_t)row * Hn + n] = acc[nt][r] + bn + pos[(size_t)l * Hn + n];
        }
    }
}

// ---------------------------------------------------------------------------
// Kernel B: fs_emb = SiLU(timestep_embedding(fs) @ w1^T + b1) @ w2^T + b2
// ---------------------------------------------------------------------------
__global__ __launch_bounds__(256)
void k_fsemb(const float* __restrict__ fs, const float* __restrict__ w1,
             const float* __restrict__ b1, const float* __restrict__ w2,
             const float* __restrict__ b2, float* __restrict__ fse)
{
    __shared__ float tf[FREQn];
    __shared__ float h1[Hn];

    const int b = blockIdx.x;
    const int h = threadIdx.x;
    const float t = fs[b];

    if (h < FREQn / 2) {
        const float fr  = expf(-logf(10000.f) * (float)h / (float)(FREQn / 2));
        const float arg = t * fr;
        tf[h]             = cosf(arg);
        tf[h + FREQn / 2] = sinf(arg);
    }
    __syncthreads();

    float acc = b1[h];
    const float* w1r = w1 + (size_t)h * FREQn;
    for (int f = 0; f < FREQn; ++f) acc += tf[f] * w1r[f];
    h1[h] = acc / (1.f + expf(-acc));       // SiLU
    __syncthreads();

    float o = b2[h];
    const float* w2r = w2 + (size_t)h * Hn;
    for (int k = 0; k < Hn; ++k) o += h1[k] * w2r[k];
    fse[(size_t)b * Hn + h] = o;
}

// ---------------------------------------------------------------------------
// Kernel C: per-row masking pipeline. One 1024-thread WG per batch row.
// Stable argsort via 64-bit (value,index) bitonic sort in LDS, prefix re-sort,
// and scattered outputs (ids_restore is the inverse permutation).
// ---------------------------------------------------------------------------
__global__ __launch_bounds__(1024)
void k_mask(const int* __restrict__ sample_ids, const int* __restrict__ eff_lens,
            const float* __restrict__ noise, int* __restrict__ ws_shuf,
            int* __restrict__ ws_lk, float* __restrict__ out_mae,
            float* __restrict__ out_restore, float* __restrict__ out_unmask)
{
    __shared__ unsigned long long key[Ln];  // 32 KB
    __shared__ int sid[Ln];                 // 16 KB
    __shared__ int nsamp;

    const int b   = blockIdx.x;
    const int tid = threadIdx.x;
    const int eff = eff_lens[b];

    if (tid == 0) nsamp = 0;
    for (int j = tid; j < Ln; j += 1024)
        sid[j] = sample_ids[(size_t)b * Ln + j];
    __syncthreads();

    int local = 0;
    for (int j = tid; j < Ln; j += 1024) {
        const bool valid = j < eff;
        const int  s  = sid[j];
        const int  p1 = (j >= 1) ? sid[j - 1] : -1;
        const int  p2 = (j >= 2) ? sid[j - 2] : -1;
        const bool segs      = valid && (s != p1);
        const bool segs_prev = (j >= 1) && ((j - 1) < eff) && (p1 != p2);
        const bool fst       = valid && segs_prev;
        float nz;
        if (!valid)               nz = __int_as_float(0x7F800000); // +inf
        else if (segs || fst)     nz = 0.f;
        else                      nz = noise[(size_t)b * Ln + j];
        const unsigned int ord = __float_as_uint(nz);   // all values >= 0
        key[j] = ((unsigned long long)ord << 32) | (unsigned int)j;
        local += segs ? 1 : 0;
    }
    atomicAdd(&nsamp, local);
    __syncthreads();

    const int ns = nsamp;
    const int lk = 2 * ns + (int)truncf((float)(eff - 2 * ns) * 0.25f);

    // Bitonic sort of (value,index) keys -> stable argsort.
    for (int k = 2; k <= Ln; k <<= 1)
        for (int jj = k >> 1; jj > 0; jj >>= 1) {
            __syncthreads();
            for (int t = tid; t < Ln; t += 1024) {
                const int ixj = t ^ jj;
                if (ixj > t) {
                    unsigned long long x = key[t], y = key[ixj];
                    const bool up = ((t & k) == 0);
                    if ((x > y) == up) { key[t] = y; key[ixj] = x; }
                }
            }
        }
    __syncthreads();

    // Extract indices; alias key region for (shuf, srt) u32 arrays.
    unsigned int myidx[Ln / 1024];
    int c = 0;
    for (int t = tid; t < Ln; t += 1024)
        myidx[c++] = (unsigned int)(key[t] & 0xFFFFFFFFu);
    __syncthreads();
    int*          shuf = (int*)key;                 // [0 .. Ln)
    unsigned int* srt  = ((unsigned int*)key) + Ln; // [Ln .. 2Ln)
    c = 0;
    for (int t = tid; t < Ln; t += 1024) {
        const unsigned int id = myidx[c++];
        shuf[t] = (int)id;
        srt[t]  = (t < lk) ? id : 0xFFFFFFFFu;      // sentinel sorts to the end
    }

    // Bitonic sort of the kept prefix indices.
    for (int k = 2; k <= Ln; k <<= 1)
        for (int jj = k >> 1; jj > 0; jj >>= 1) {
            __syncthreads();
            for (int t = tid; t < Ln; t += 1024) {
                const int ixj = t ^ jj;
                if (ixj > t) {
                    unsigned int x = srt[t], y = srt[ixj];
                    const bool up = ((t & k) == 0);
                    if ((x > y) == up) { srt[t] = y; srt[ixj] = x; }
                }
            }
        }
    __syncthreads();

    for (int t = tid; t < Ln; t += 1024) {
        const int id = (t < lk) ? (int)srt[t] : shuf[t];
        ws_shuf[(size_t)b * Ln + t] = id;
        out_restore[(size_t)b * Ln + id] = (float)t;           // inverse perm
        out_mae[(size_t)b * Ln + id] = (t >= lk && t < eff) ? 1.f : 0.f;
        out_unmask[(size_t)b * Ln + t] = (t < lk) ? (float)sid[id] : -1.f;
    }
    if (tid == 0) ws_lk[b] = lk;
}

// ---------------------------------------------------------------------------
// Kernel D: seq[b,pos,:] = pos < len_keep ? emb[b, shuf[b,pos], :] + fse[b,:] : 0
// ---------------------------------------------------------------------------
__global__ __launch_bounds__(256)
void k_gather(const float* __restrict__ emb, const float* __restrict__ fse,
              const int* __restrict__ shuf, const int* __restrict__ lk,
              float* __restrict__ out_seq)
{
    const int tid  = threadIdx.x;
    const int pos4 = blockIdx.x * 4 + (tid >> 6);
    const int b    = pos4 >> 12;            // / Ln
    const int pos  = pos4 & (Ln - 1);
    const int h    = (tid & 63) * 4;
    const int keepN = lk[b];

    float4 o;
    if (pos < keepN) {
        const int id = shuf[(size_t)b * Ln + pos];
        const float4 e = *(const float4*)(emb + ((size_t)b * Ln + id) * Hn + h);
        const float4 f = *(const float4*)(fse + (size_t)b * Hn + h);
        o = make_float4(e.x + f.x, e.y + f.y, e.z + f.z, e.w + f.w);
    } else {
        o = make_float4(0.f, 0.f, 0.f, 0.f);
    }
    *(float4*)(out_seq + ((size_t)b * Ln + pos) * Hn + h) = o;
}

// ---------------------------------------------------------------------------
extern "C" void kernel_launch(void* const* d_in, const int* in_sizes, int n_in,
                              void* d_out, int out_size, void* d_ws, size_t ws_size,
                              hipStream_t stream)
{
    (void)in_sizes; (void)n_in; (void)out_size; (void)ws_size;

    const float* patches    = (const float*)d_in[0];
    const int*   sample_ids = (const int*)  d_in[1];
    const int*   eff_lens   = (const int*)  d_in[2];
    const float* noise      = (const float*)d_in[3];
    const float* fs         = (const float*)d_in[4];
    const float* W          = (const float*)d_in[5];
    const float* bias       = (const float*)d_in[6];
    const float* pos        = (const float*)d_in[7];
    const float* w1         = (const float*)d_in[8];
    const float* b1         = (const float*)d_in[9];
    const float* w2         = (const float*)d_in[10];
    const float* b2         = (const float*)d_in[11];

    // workspace layout
    float* emb  = (float*)d_ws;                            // B*L*H
    float* fse  = emb + (size_t)Bn * Ln * Hn;              // B*H
    int*   shuf = (int*)(fse + (size_t)Bn * Hn);           // B*L
    int*   lkp  = shuf + (size_t)Bn * Ln;                  // B
    float* WtG2 = (float*)(lkp + 64);                      // PD*H pair-interleaved W

    // output layout (tuple concatenated flat, f32)
    float* out_seq     = (float*)d_out;                    // B*L*H
    float* out_mae     = out_seq + (size_t)Bn * Ln * Hn;   // B*L
    float* out_restore = out_mae + (size_t)Bn * Ln;        // B*L
    float* out_unmask  = out_restore + (size_t)Bn * Ln;    // B*L

    k_wtr   <<<PDn, Hn, 0, stream>>>(W, WtG2);
    k_proj  <<<dim3(Hn / 64, (Bn * Ln) / 64), 128, 0, stream>>>(patches, WtG2, bias, pos, emb);
    k_fsemb <<<Bn, 256, 0, stream>>>(fs, w1, b1, w2, b2, fse);
    k_mask  <<<Bn, 1024, 0, stream>>>(sample_ids, eff_lens, noise, shuf, lkp,
                                      out_mae, out_restore, out_unmask);
    k_gather<<<(Bn * Ln) / 4, 256, 0, stream>>>(emb, fse, shuf, lkp, out_seq);
}